// GraphTransformer_11828339933760
// MI455X (gfx1250) — compile-verified
//
#include <hip/hip_runtime.h>
#include <math.h>

typedef __attribute__((ext_vector_type(2))) float v2f;
typedef __attribute__((ext_vector_type(8))) float v8f;

#define FIN 256
#define FH  128
#define FC  40

// ---------------- degree / norm ----------------

__global__ void k_deg_init(float* __restrict__ deg, int N) {
    int i = blockIdx.x * blockDim.x + threadIdx.x;
    if (i < N) deg[i] = 1.0f;  // self-loop contribution
}

__global__ void k_deg_edges(float* __restrict__ deg, const long long* __restrict__ ei, int E) {
    int e = blockIdx.x * blockDim.x + threadIdx.x;
    if (e < E) {
        long long c = ei[(size_t)E + e];   // col = edge_index[1][e]
        atomicAdd(&deg[c], 1.0f);
    }
}

__global__ void k_dinv(float* __restrict__ deg, int N) {
    int i = blockIdx.x * blockDim.x + threadIdx.x;
    if (i < N) deg[i] = rsqrtf(deg[i]);    // deg >= 1 always (self loops)
}

// ---------------- GEMM1: h = x @ W1  [N,256]x[256,128], fp32 WMMA ----------------

__global__ __launch_bounds__(256)
void k_gemm1(const float* __restrict__ x, const float* __restrict__ W1,
             float* __restrict__ h, int N) {
    const int lane = threadIdx.x & 31;
    const int wv   = threadIdx.x >> 5;     // 8 waves: one 16-col tile each
    const int lid  = lane & 15;
    const int hi   = lane >> 4;            // 0: K={k,k+1}, 1: K={k+2,k+3}
    const int rowBase = blockIdx.x * 16;
    const int nBase   = wv * 16;

    int rowA = rowBase + lid;
    if (rowA >= N) rowA = N - 1;           // clamp (keeps EXEC all-1s)
    const float* xr = x + (size_t)rowA * FIN;
    const float* wc = W1 + nBase + lid;

    v8f acc = {};
#pragma unroll
    for (int k0 = 0; k0 < FIN; k0 += 4) {
        const int ka = k0 + hi * 2;
        v2f a, b;
        a.x = xr[ka];
        a.y = xr[ka + 1];
        b.x = wc[(size_t)ka * FH];
        b.y = wc[(size_t)(ka + 1) * FH];
        acc = __builtin_amdgcn_wmma_f32_16x16x4_f32(false, a, false, b,
                                                    (short)0, acc, false, false);
    }
#pragma unroll
    for (int r = 0; r < 8; ++r) {
        int m = rowBase + r + hi * 8;
        if (m < N) h[(size_t)m * FH + nBase + lid] = acc[r];
    }
}

// ---------------- layer-1 aggregation ----------------

// init accumulator with self-loop message: agg[i][f] = hlin[i][f] * dinv[i]^2
__global__ void k_self1(const float* __restrict__ hlin, float* __restrict__ agg,
                        const float* __restrict__ dinv, int N) {
    size_t idx = (size_t)blockIdx.x * blockDim.x + threadIdx.x;
    if (idx < (size_t)N * FH) {
        int i = (int)(idx >> 7);
        float d = dinv[i];
        agg[idx] = hlin[idx] * d * d;
    }
}

// one wave32 per edge; each lane moves float4 (4*32 = 128 features)
__global__ __launch_bounds__(256)
void k_scatter1(const float* __restrict__ hlin, float* __restrict__ agg,
                const long long* __restrict__ ei, const float* __restrict__ dinv, int E) {
    int wave = (int)(((size_t)blockIdx.x * blockDim.x + threadIdx.x) >> 5);
    int lane = threadIdx.x & 31;
    if (wave >= E) return;
    long long r = ei[wave];
    long long c = ei[(size_t)E + wave];
    float norm = dinv[r] * dinv[c];
    const float4 v = *(const float4*)(hlin + (size_t)r * FH + lane * 4);
    float* dst = agg + (size_t)c * FH + lane * 4;
    atomicAdd(dst + 0, v.x * norm);
    atomicAdd(dst + 1, v.y * norm);
    atomicAdd(dst + 2, v.z * norm);
    atomicAdd(dst + 3, v.w * norm);
}

__global__ void k_bias_relu1(float* __restrict__ agg, const float* __restrict__ b1, int N) {
    size_t idx = (size_t)blockIdx.x * blockDim.x + threadIdx.x;
    if (idx < (size_t)N * FH) {
        float v = agg[idx] + b1[idx & (FH - 1)];
        agg[idx] = v > 0.0f ? v : 0.0f;
    }
}

// ---------------- GEMM2: h2 = relu_agg @ W2  [N,128]x[128,40], 3 padded col tiles ----------------

__global__ __launch_bounds__(96)
void k_gemm2(const float* __restrict__ h, const float* __restrict__ W2,
             float* __restrict__ o, int N) {
    const int lane = threadIdx.x & 31;
    const int wv   = threadIdx.x >> 5;     // 3 waves: col tiles 0..2 (48 padded cols)
    const int lid  = lane & 15;
    const int hi   = lane >> 4;
    const int rowBase = blockIdx.x * 16;
    const int nBase   = wv * 16;
    const int n = nBase + lid;
    const bool nv = (n < FC);

    int rowA = rowBase + lid;
    if (rowA >= N) rowA = N - 1;
    const float* hr = h + (size_t)rowA * FH;

    v8f acc = {};
#pragma unroll
    for (int k0 = 0; k0 < FH; k0 += 4) {
        const int ka = k0 + hi * 2;
        v2f a, b;
        a.x = hr[ka];
        a.y = hr[ka + 1];
        b.x = nv ? W2[(size_t)ka * FC + n] : 0.0f;       // zero-pad cols 40..47
        b.y = nv ? W2[(size_t)(ka + 1) * FC + n] : 0.0f;
        acc = __builtin_amdgcn_wmma_f32_16x16x4_f32(false, a, false, b,
                                                    (short)0, acc, false, false);
    }
#pragma unroll
    for (int r = 0; r < 8; ++r) {
        int m = rowBase + r + hi * 8;
        if (m < N && nv) o[(size_t)m * FC + n] = acc[r];
    }
}

// ---------------- layer-2 aggregation + log_softmax ----------------

// out[i][c] = h2[i][c]*dinv[i]^2 + b2[c]  (self-loop + bias folded into init)
__global__ void k_self2(const float* __restrict__ h2, float* __restrict__ out,
                        const float* __restrict__ dinv, const float* __restrict__ b2, int N) {
    size_t idx = (size_t)blockIdx.x * blockDim.x + threadIdx.x;
    if (idx < (size_t)N * FC) {
        int i = (int)(idx / FC);
        int c = (int)(idx - (size_t)i * FC);
        float d = dinv[i];
        out[idx] = h2[idx] * d * d + b2[c];
    }
}

// one wave32 per edge; lanes 0..31 cover c=lane, lanes 0..7 also cover c=lane+32
__global__ __launch_bounds__(256)
void k_scatter2(const float* __restrict__ h2, float* __restrict__ out,
                const long long* __restrict__ ei, const float* __restrict__ dinv, int E) {
    int wave = (int)(((size_t)blockIdx.x * blockDim.x + threadIdx.x) >> 5);
    int lane = threadIdx.x & 31;
    if (wave >= E) return;
    long long r = ei[wave];
    long long c = ei[(size_t)E + wave];
    float norm = dinv[r] * dinv[c];
    const float* src = h2 + (size_t)r * FC;
    float* dst = out + (size_t)c * FC;
    atomicAdd(dst + lane, src[lane] * norm);
    if (lane < FC - 32) atomicAdd(dst + 32 + lane, src[32 + lane] * norm);
}

// one wave32 per node, in-place log_softmax over 40 values
__global__ __launch_bounds__(256)
void k_logsoftmax(float* __restrict__ out, int N) {
    int node = blockIdx.x * 8 + (threadIdx.x >> 5);
    int lane = threadIdx.x & 31;
    if (node >= N) return;
    float* p = out + (size_t)node * FC;
    float v0 = p[lane];                                       // c = 0..31
    float v1 = (lane < FC - 32) ? p[32 + lane] : -__builtin_inff();
    float m = fmaxf(v0, v1);
#pragma unroll
    for (int off = 16; off > 0; off >>= 1)
        m = fmaxf(m, __shfl_xor(m, off, 32));
    float s = __expf(v0 - m) + ((lane < FC - 32) ? __expf(v1 - m) : 0.0f);
#pragma unroll
    for (int off = 16; off > 0; off >>= 1)
        s += __shfl_xor(s, off, 32);
    float lse = m + __logf(s);
    p[lane] = v0 - lse;
    if (lane < FC - 32) p[32 + lane] = v1 - lse;
}

// ---------------- launch ----------------

extern "C" void kernel_launch(void* const* d_in, const int* in_sizes, int n_in,
                              void* d_out, int out_size, void* d_ws, size_t ws_size,
                              hipStream_t stream) {
    const float*     x  = (const float*)d_in[0];
    const long long* ei = (const long long*)d_in[1];
    const float*     W1 = (const float*)d_in[2];
    const float*     b1 = (const float*)d_in[3];
    const float*     W2 = (const float*)d_in[4];
    const float*     b2 = (const float*)d_in[5];
    float* out = (float*)d_out;

    const int N = in_sizes[0] / FIN;   // 100000
    const int E = in_sizes[1] / 2;     // 1600000

    float* ws   = (float*)d_ws;
    float* dinv = ws;                           // N floats (deg -> dinv in place)
    float* bufA = ws + N;                       // N*FH: layer1 linear, reused as layer2 linear (N*FC)
    float* bufB = bufA + (size_t)N * FH;        // N*FH: layer1 aggregated

    const int tpb = 256;
    // degrees + dinv
    k_deg_init <<<(N + tpb - 1) / tpb, tpb, 0, stream>>>(dinv, N);
    k_deg_edges<<<(E + tpb - 1) / tpb, tpb, 0, stream>>>(dinv, ei, E);
    k_dinv     <<<(N + tpb - 1) / tpb, tpb, 0, stream>>>(dinv, N);
    // layer 1
    k_gemm1    <<<(N + 15) / 16, 256, 0, stream>>>(x, W1, bufA, N);
    k_self1    <<<(int)(((size_t)N * FH + tpb - 1) / tpb), tpb, 0, stream>>>(bufA, bufB, dinv, N);
    k_scatter1 <<<(E + 7) / 8, 256, 0, stream>>>(bufA, bufB, ei, dinv, E);
    k_bias_relu1<<<(int)(((size_t)N * FH + tpb - 1) / tpb), tpb, 0, stream>>>(bufB, b1, N);
    // layer 2
    k_gemm2    <<<(N + 15) / 16, 96, 0, stream>>>(bufB, W2, bufA, N);
    k_self2    <<<(int)(((size_t)N * FC + tpb - 1) / tpb), tpb, 0, stream>>>(bufA, out, dinv, b2, N);
    k_scatter2 <<<(E + 7) / 8, 256, 0, stream>>>(bufA, out, ei, dinv, E);
    k_logsoftmax<<<(N + 7) / 8, 256, 0, stream>>>(out, N);
}